// AED_42013370090080
// MI455X (gfx1250) — compile-verified
//
#include <hip/hip_runtime.h>
#include <math.h>

// ---------------------------------------------------------------------------
// AED distillation loss, fused single-pass for MI455X (gfx1250, wave32).
//   d_out[0]        = mean_b L[b]
//   d_out[1 .. B*C] = mimic = mean of the 4 logit tensors
// One workgroup per row (B blocks x 256 threads = 8 wave32).
// Traffic: 512MB read + 128MB write once => ~27us at 23.3 TB/s. Main loop
// streams tiles through LDS with GLOBAL_LOAD_ASYNC_TO_LDS_B128 (ASYNCcnt
// pipeline, lane-private tiles => no barriers), falling back to direct
// nontemporal b128 loads if the async builtin is unavailable.
// ---------------------------------------------------------------------------

typedef float v4f __attribute__((ext_vector_type(4)));
typedef float v2f __attribute__((ext_vector_type(2)));
typedef float v8f __attribute__((ext_vector_type(8)));
typedef int   v4i __attribute__((ext_vector_type(4)));

#define TDIST     3.0f
#define INV_TDIST (1.0f / 3.0f)

#if defined(__has_builtin)
#if __has_builtin(__builtin_amdgcn_global_load_async_to_lds_b128)
#define HAVE_ASYNC_LDS 1
#endif
#endif

#define TILE_F4 256   // float4 per model per tile (1024 floats = 4KB)
#define NBUF    3     // triple buffer: 4 models * 3 * 4KB = 48KB LDS

__device__ __forceinline__ void wait_async_le(int n) {
#if defined(HAVE_ASYNC_LDS)
#if __has_builtin(__builtin_amdgcn_s_wait_asynccnt)
  if (n >= 8)      __builtin_amdgcn_s_wait_asynccnt(8);
  else if (n >= 4) __builtin_amdgcn_s_wait_asynccnt(4);
  else             __builtin_amdgcn_s_wait_asynccnt(0);
#else
  if (n >= 8)      asm volatile("s_wait_asynccnt 0x8" ::: "memory");
  else if (n >= 4) asm volatile("s_wait_asynccnt 0x4" ::: "memory");
  else             asm volatile("s_wait_asynccnt 0x0" ::: "memory");
#endif
#else
  (void)n;
#endif
}

// Per-model online accumulators:
//  ms = max(o/T), ss = sum exp(o/T - ms), s1 = sum exp(o - 3*ms)  [= sum e_s^3]
//  mt = max(t/T), st = sum exp(t/T - mt), w  = sum exp(t/T - mt)*(mimic - o)
struct SAcc { float ms, ss, s1, mt, st, w; };

__device__ __forceinline__ void upd(SAcc& a, float o, float mim) {
  float xs = o * INV_TDIST;
  if (xs > a.ms) {                       // rare rescale branch
    float c = __expf(a.ms - xs);
    a.ss *= c; a.s1 *= c * c * c; a.ms = xs;
  }
  float e = __expf(xs - a.ms);
  a.ss += e; a.s1 += e * e * e;          // e^3 == exp(o - 3*ms): scale-1 lse free
  float xt = (0.8f * o + 0.2f * mim) * INV_TDIST;
  if (xt > a.mt) {
    float c = __expf(a.mt - xt);
    a.st *= c; a.w *= c; a.mt = xt;
  }
  float et = __expf(xt - a.mt);
  a.st += et; a.w += et * (mim - o);
}

__device__ __forceinline__ void mergeAcc(SAcc& a, const SAcc& b) {
  float M  = fmaxf(a.ms, b.ms);
  float ca = __expf(a.ms - M), cb = __expf(b.ms - M);
  a.ss = a.ss * ca + b.ss * cb;
  a.s1 = a.s1 * ca * ca * ca + b.s1 * cb * cb * cb;
  a.ms = M;
  M  = fmaxf(a.mt, b.mt);
  ca = __expf(a.mt - M); cb = __expf(b.mt - M);
  a.st = a.st * ca + b.st * cb;
  a.w  = a.w  * ca + b.w  * cb;
  a.mt = M;
}

__device__ __forceinline__ SAcc shflAcc(const SAcc& a, int m) {
  SAcc r;
  r.ms = __shfl_xor(a.ms, m); r.ss = __shfl_xor(a.ss, m); r.s1 = __shfl_xor(a.s1, m);
  r.mt = __shfl_xor(a.mt, m); r.st = __shfl_xor(a.st, m); r.w  = __shfl_xor(a.w,  m);
  return r;
}

// One float4 chunk (per model) through all accumulators + mimic store.
__device__ __forceinline__ void processChunk(
    v4f a0, v4f a1, v4f a2, v4f a3, int i,
    float* __restrict__ mout, SAcc* acc, float* d) {
  v4f mim = 0.25f * (a0 + a1 + a2 + a3);
#pragma unroll
  for (int k = 0; k < 4; ++k)
    __builtin_nontemporal_store(mim[k], mout + 4 * i + k);  // +1 float offset => scalar
#pragma unroll
  for (int k = 0; k < 4; ++k) {
    float f0 = a0[k], f1 = a1[k], f2 = a2[k], f3 = a3[k], fm = mim[k];
    upd(acc[0], f0, fm); upd(acc[1], f1, fm);
    upd(acc[2], f2, fm); upd(acc[3], f3, fm);
    d[0] += f0 * f0; d[1] += f0 * f1; d[2] += f0 * f2; d[3] += f0 * f3;
    d[4] += f1 * f1; d[5] += f1 * f2; d[6] += f1 * f3;
    d[7] += f2 * f2; d[8] += f2 * f3; d[9] += f3 * f3;
  }
}

__global__ __launch_bounds__(256) void aed_row_kernel(
    const float* __restrict__ o0p, const float* __restrict__ o1p,
    const float* __restrict__ o2p, const float* __restrict__ o3p,
    const long long* __restrict__ tgt,
    float* __restrict__ out,       // out[0] reserved for loss; out+1 = mimic
    float* __restrict__ rowloss,   // [B] scratch
    int C) {
  const int b = blockIdx.x;
  const int tid = threadIdx.x;
  const int nthreads = blockDim.x;         // 256 = 8 wave32
  const size_t base = (size_t)b * (size_t)C;

  const v4f* __restrict__ p0 = (const v4f*)(o0p + base);
  const v4f* __restrict__ p1 = (const v4f*)(o1p + base);
  const v4f* __restrict__ p2 = (const v4f*)(o2p + base);
  const v4f* __restrict__ p3 = (const v4f*)(o3p + base);
  float* __restrict__ mout = out + 1 + base;

  SAcc acc[4];
#pragma unroll
  for (int j = 0; j < 4; ++j) {
    acc[j].ms = -INFINITY; acc[j].ss = 0.f; acc[j].s1 = 0.f;
    acc[j].mt = -INFINITY; acc[j].st = 0.f; acc[j].w  = 0.f;
  }
  float d[10];
#pragma unroll
  for (int q = 0; q < 10; ++q) d[q] = 0.f;

  const int C4 = C >> 2;

#if defined(HAVE_ASYNC_LDS)
  // ---- async global->LDS pipeline (lane-private tiles, no barriers) ----
  // Builtin wants non-const int4 pointers: AS1 (__device__) src, AS3 dst.
  typedef __attribute__((address_space(1))) v4i* g_v4i;
  typedef __attribute__((address_space(3))) v4i* l_v4i;
  __shared__ v4f tile[4][NBUF][TILE_F4];
  const int T = C4 / TILE_F4;              // 31 full tiles for C=32000

  auto issue = [&](int p) {
    const int buf = p % NBUF;
    const int gi = p * TILE_F4 + tid;
    __builtin_amdgcn_global_load_async_to_lds_b128(
        (g_v4i)(p0 + gi), (l_v4i)&tile[0][buf][tid], 0, 0);
    __builtin_amdgcn_global_load_async_to_lds_b128(
        (g_v4i)(p1 + gi), (l_v4i)&tile[1][buf][tid], 0, 0);
    __builtin_amdgcn_global_load_async_to_lds_b128(
        (g_v4i)(p2 + gi), (l_v4i)&tile[2][buf][tid], 0, 0);
    __builtin_amdgcn_global_load_async_to_lds_b128(
        (g_v4i)(p3 + gi), (l_v4i)&tile[3][buf][tid], 0, 0);
  };

  if (T > 0) issue(0);
  if (T > 1) issue(1);
  for (int t = 0; t < T; ++t) {
    if (t + 2 < T) issue(t + 2);
    int newer = T - 1 - t;                 // tiles still in flight after tile t
    wait_async_le(4 * (newer > 2 ? 2 : newer));
    const int buf = t % NBUF;
    v4f a0 = tile[0][buf][tid];            // ds_load_b128, lane-private slot
    v4f a1 = tile[1][buf][tid];
    v4f a2 = tile[2][buf][tid];
    v4f a3 = tile[3][buf][tid];
    processChunk(a0, a1, a2, a3, t * TILE_F4 + tid, mout, acc, d);
  }
  // remainder (64 float4 for C=32000): direct nontemporal loads
  for (int i = T * TILE_F4 + tid; i < C4; i += nthreads) {
    v4f a0 = __builtin_nontemporal_load(p0 + i);
    v4f a1 = __builtin_nontemporal_load(p1 + i);
    v4f a2 = __builtin_nontemporal_load(p2 + i);
    v4f a3 = __builtin_nontemporal_load(p3 + i);
    processChunk(a0, a1, a2, a3, i, mout, acc, d);
  }
#else
  // ---- fallback: direct streaming loads + prefetch hints ----
  for (int i = tid; i < C4; i += nthreads) {
    __builtin_prefetch((const void*)(p0 + i + 2 * nthreads), 0, 0);
    __builtin_prefetch((const void*)(p1 + i + 2 * nthreads), 0, 0);
    __builtin_prefetch((const void*)(p2 + i + 2 * nthreads), 0, 0);
    __builtin_prefetch((const void*)(p3 + i + 2 * nthreads), 0, 0);
    v4f a0 = __builtin_nontemporal_load(p0 + i);
    v4f a1 = __builtin_nontemporal_load(p1 + i);
    v4f a2 = __builtin_nontemporal_load(p2 + i);
    v4f a3 = __builtin_nontemporal_load(p3 + i);
    processChunk(a0, a1, a2, a3, i, mout, acc, d);
  }
#endif

  // ---- wave32 tree reduction ----
#pragma unroll
  for (int m = 16; m >= 1; m >>= 1) {
#pragma unroll
    for (int j = 0; j < 4; ++j) { SAcc o = shflAcc(acc[j], m); mergeAcc(acc[j], o); }
#pragma unroll
    for (int q = 0; q < 10; ++q) d[q] += __shfl_xor(d[q], m);
  }

  // ---- cross-wave merge (8 waves) ----
  __shared__ float sm[8][34];
  const int wave = tid >> 5, lane = tid & 31;
  if (lane == 0) {
#pragma unroll
    for (int j = 0; j < 4; ++j) {
      sm[wave][6 * j + 0] = acc[j].ms; sm[wave][6 * j + 1] = acc[j].ss;
      sm[wave][6 * j + 2] = acc[j].s1; sm[wave][6 * j + 3] = acc[j].mt;
      sm[wave][6 * j + 4] = acc[j].st; sm[wave][6 * j + 5] = acc[j].w;
    }
#pragma unroll
    for (int q = 0; q < 10; ++q) sm[wave][24 + q] = d[q];
  }
  __syncthreads();

  if (tid == 0) {
    const int nw = nthreads >> 5;
    for (int wv = 1; wv < nw; ++wv) {
#pragma unroll
      for (int j = 0; j < 4; ++j) {
        SAcc o;
        o.ms = sm[wv][6 * j + 0]; o.ss = sm[wv][6 * j + 1]; o.s1 = sm[wv][6 * j + 2];
        o.mt = sm[wv][6 * j + 3]; o.st = sm[wv][6 * j + 4]; o.w  = sm[wv][6 * j + 5];
        mergeAcc(acc[j], o);
      }
#pragma unroll
      for (int q = 0; q < 10; ++q) d[q] += sm[wv][24 + q];
    }

    const long long t = tgt[b];
    float ot[4];
    ot[0] = o0p[base + (size_t)t]; ot[1] = o1p[base + (size_t)t];
    ot[2] = o2p[base + (size_t)t]; ot[3] = o3p[base + (size_t)t];

    const float invC = 1.0f / (float)C;
    float CE[4], KD[4];
#pragma unroll
    for (int j = 0; j < 4; ++j) {
      float lse_s = acc[j].ms + __logf(acc[j].ss);           // lse(o/T)
      float lse1  = 3.0f * acc[j].ms + __logf(acc[j].s1);    // lse(o)
      float lse_t = acc[j].mt + __logf(acc[j].st);           // lse(teacher/T)
      CE[j] = lse1 - ot[j];
      // KD = T^2/C * [0.2/T * E_pt[mimic-o] + lse_s - lse_t]   (sum p_t = 1)
      KD[j] = (TDIST * TDIST) * invC *
              ((0.2f * INV_TDIST) * (acc[j].w / acc[j].st) + (lse_s - lse_t));
    }
    // masked Gram = streaming Gram - rank-1 target correction
    float n0 = sqrtf(d[0] - ot[0] * ot[0]);
    float n1 = sqrtf(d[4] - ot[1] * ot[1]);
    float n2 = sqrtf(d[7] - ot[2] * ot[2]);
    float n3 = sqrtf(d[9] - ot[3] * ot[3]);
    float cs = (d[1] - ot[0] * ot[1]) / (n0 * n1) +
               (d[2] - ot[0] * ot[2]) / (n0 * n2) +
               (d[3] - ot[0] * ot[3]) / (n0 * n3) +
               (d[5] - ot[1] * ot[2]) / (n1 * n2) +
               (d[6] - ot[1] * ot[3]) / (n1 * n3) +
               (d[8] - ot[2] * ot[3]) / (n2 * n3);
    float s = 0.f, mn = CE[0];
#pragma unroll
    for (int j = 0; j < 4; ++j) { s += CE[j] + KD[j]; mn = fminf(mn, CE[j]); }
    rowloss[b] = s + 0.7f * mn + 0.3f * cs;
  }
}

// ---------------------------------------------------------------------------
// Final 1024 -> 1 reduction. One wave32, V_WMMA_F32_16X16X4_F32 with B=ones:
// D[m][n] = sum_k A[m][k], so sum_m D[m][0] = sum of all 64 A slots — correct
// for any (m,k)<->(lane,vgpr) mapping. D column 0 = lanes 0 (M=0..7) and
// 16 (M=8..15) per the 16x16 f32 C/D layout.
// ---------------------------------------------------------------------------
__global__ __launch_bounds__(32) void aed_final_kernel(
    const float* __restrict__ rl, float* __restrict__ out, int B) {
  const int lane = threadIdx.x;
  v2f a; a.x = 0.f; a.y = 0.f;
  for (int i = lane; i < B; i += 64)       a.x += rl[i];
  for (int i = lane + 32; i < B; i += 64)  a.y += rl[i];

  float total;
#if __has_builtin(__builtin_amdgcn_wmma_f32_16x16x4_f32)
  v2f bones; bones.x = 1.0f; bones.y = 1.0f;
  v8f c = {};
  v8f dd = __builtin_amdgcn_wmma_f32_16x16x4_f32(
      false, a, false, bones, (short)0, c, false, false);
  float p = dd[0] + dd[1] + dd[2] + dd[3] + dd[4] + dd[5] + dd[6] + dd[7];
  total = __shfl(p, 0) + __shfl(p, 16);
#else
  float p = a.x + a.y;
#pragma unroll
  for (int m = 16; m >= 1; m >>= 1) p += __shfl_xor(p, m);
  total = p;
#endif
  if (lane == 0) out[0] = total / (float)B;
}

extern "C" void kernel_launch(void* const* d_in, const int* in_sizes, int n_in,
                              void* d_out, int out_size, void* d_ws, size_t ws_size,
                              hipStream_t stream) {
  (void)n_in; (void)out_size; (void)ws_size;
  const int B = in_sizes[4];               // 1024 targets
  const int C = in_sizes[0] / B;           // 32000
  const float* o0 = (const float*)d_in[0];
  const float* o1 = (const float*)d_in[1];
  const float* o2 = (const float*)d_in[2];
  const float* o3 = (const float*)d_in[3];
  const long long* tg = (const long long*)d_in[4];
  float* out = (float*)d_out;
  float* rl  = (float*)d_ws;               // B floats of scratch

  aed_row_kernel<<<B, 256, 0, stream>>>(o0, o1, o2, o3, tg, out, rl, C);
  aed_final_kernel<<<1, 32, 0, stream>>>(rl, out, B);
}